// PairwiseDotproducts_21088289423831
// MI455X (gfx1250) — compile-verified
//
#include <hip/hip_runtime.h>

// ---------------------------------------------------------------------------
// PairwiseDotproducts on MI455X (gfx1250, wave32)
//
// triu(X Xᵀ, 1).sum() == ( ||Σ_i x_i||² − ||X||_F² ) / 2
//
// -> one streaming pass over X (33.5 MB ≈ 1.4 µs @ 23.3 TB/s), then a tiny
//    s·s self-dot done with V_WMMA_F32_16X16X4_F32 (trace of segment Gram).
// ---------------------------------------------------------------------------

typedef __attribute__((ext_vector_type(2))) float v2f;
typedef __attribute__((ext_vector_type(8))) float v8f;

#define KCOLS 1024
#define NROWS 8192
#define NB1   256                       // stage-1 blocks
#define ROWS_PER_B1 (NROWS / NB1)       // 32 rows per block
#define NB2   8                         // stage-2 blocks
#define COLS_PER_B2 (KCOLS / NB2)       // 128 cols per block

// ws layout (floats):
//   P  [NB1*KCOLS] : per-block column partial sums   (offset 0)
//   SS [NB1]       : per-block sum-of-squares        (offset NB1*KCOLS)
//   D  [NB2]       : per-block s·s partials          (offset NB1*KCOLS+NB1)

// ---------------------------------------------------------------------------
// Stage 1: stream X once with b128 loads. Block b owns rows [b*32, b*32+32).
// Thread t owns columns 4t..4t+3.
// ---------------------------------------------------------------------------
__global__ __launch_bounds__(256) void pd_stage1(const float4* __restrict__ x4,
                                                 float4* __restrict__ P4,
                                                 float* __restrict__ SS) {
  const int b = blockIdx.x;
  const int t = threadIdx.x;
  const float4* row = x4 + (size_t)b * ROWS_PER_B1 * (KCOLS / 4) + t;

  float ax = 0.f, ay = 0.f, az = 0.f, aw = 0.f;
  float ss = 0.f;
#pragma unroll
  for (int r = 0; r < ROWS_PER_B1; ++r) {
    float4 v = row[(size_t)r * (KCOLS / 4)];   // global_load_b128, coalesced
    ax += v.x; ay += v.y; az += v.z; aw += v.w;
    ss = fmaf(v.x, v.x, ss);
    ss = fmaf(v.y, v.y, ss);
    ss = fmaf(v.z, v.z, ss);
    ss = fmaf(v.w, v.w, ss);
  }
  float4 acc; acc.x = ax; acc.y = ay; acc.z = az; acc.w = aw;
  P4[(size_t)b * (KCOLS / 4) + t] = acc;       // global_store_b128

  __shared__ float red[256];
  red[t] = ss;
  __syncthreads();
#pragma unroll
  for (int s = 128; s > 0; s >>= 1) {
    if (t < s) red[t] += red[t + s];
    __syncthreads();
  }
  if (t == 0) SS[b] = red[0];
}

// ---------------------------------------------------------------------------
// Stage 2: block j reduces P over the block dim for columns
// [j*128, j*128+128) -> s chunk in LDS; wave 0 computes the chunk's
// contribution to s·s = trace(SegGram) with two f32 WMMAs.
//
// A (16x4 f32) ISA layout: lanes 0-15 hold M=lane with VGPR0=K0,VGPR1=K1;
// lanes 16-31 hold M=lane-16 with VGPR0=K2,VGPR1=K3.  With A[m][k] =
// s[64*chunk + 4m + k] and B = Aᵀ-compatible (B[k][n] = s[64*chunk + 4n + k]),
// every lane's A and B register pair is the identical float2 from LDS.
// D[m][n] = seg_m · seg_n; trace(D) = Σ_k s_k².
// ---------------------------------------------------------------------------
__global__ __launch_bounds__(128) void pd_stage2(const float* __restrict__ P,
                                                 float* __restrict__ D) {
  const int j = blockIdx.x;
  const int t = threadIdx.x;
  const int col = j * COLS_PER_B2 + t;

  float s = 0.f;
#pragma unroll 8
  for (int b = 0; b < NB1; ++b) s += P[(size_t)b * KCOLS + col];

  __shared__ float sbuf[COLS_PER_B2];
  __shared__ float tred[32];
  sbuf[t] = s;
  __syncthreads();

  if (t < 32) {                       // wave 0 fully active -> EXEC all 1s
    const int m    = t & 15;
    const int half = t >> 4;
    v8f c = {};                       // fp32 16x16 accumulator (8 VGPRs)
#pragma unroll
    for (int chunk = 0; chunk < COLS_PER_B2 / 64; ++chunk) {
      const float* base = sbuf + chunk * 64 + m * 4 + half * 2;
      v2f a;
      a.x = base[0];
      a.y = base[1];
      // D = A*Aᵀ(seg) + C   -> v_wmma_f32_16x16x4_f32
      c = __builtin_amdgcn_wmma_f32_16x16x4_f32(
              /*neg_a=*/false, a, /*neg_b=*/false, a,
              /*c_mod=*/(short)0, c, /*reuse_a=*/false, /*reuse_b=*/false);
    }
    // Diagonal of D: VGPR v @ lane v (M=v) and VGPR v @ lane 24+v (M=v+8).
    float contrib = 0.f;
#pragma unroll
    for (int v = 0; v < 8; ++v) {
      if (t == v || t == 24 + v) contrib += c[v];
    }
    tred[t] = contrib;
  }
  __syncthreads();
  if (t == 0) {
    float tr = 0.f;
#pragma unroll
    for (int i = 0; i < 32; ++i) tr += tred[i];
    D[j] = tr;
  }
}

// ---------------------------------------------------------------------------
// Stage 3: out = 0.5 * ( Σ D[j]  −  Σ SS[b] )
// ---------------------------------------------------------------------------
__global__ __launch_bounds__(256) void pd_stage3(const float* __restrict__ SS,
                                                 const float* __restrict__ D,
                                                 float* __restrict__ out) {
  const int t = threadIdx.x;
  __shared__ float red[256];
  red[t] = SS[t];                     // NB1 == 256
  __syncthreads();
#pragma unroll
  for (int s = 128; s > 0; s >>= 1) {
    if (t < s) red[t] += red[t + s];
    __syncthreads();
  }
  if (t == 0) {
    float dot = 0.f;
#pragma unroll
    for (int i = 0; i < NB2; ++i) dot += D[i];
    out[0] = 0.5f * (dot - red[0]);
  }
}

extern "C" void kernel_launch(void* const* d_in, const int* in_sizes, int n_in,
                              void* d_out, int out_size, void* d_ws, size_t ws_size,
                              hipStream_t stream) {
  (void)in_sizes; (void)n_in; (void)out_size; (void)ws_size;

  const float* x = (const float*)d_in[0];          // 8192 x 1024 fp32
  float* ws = (float*)d_ws;
  float* P  = ws;                                  // NB1*KCOLS floats (1 MB)
  float* SS = ws + (size_t)NB1 * KCOLS;            // NB1 floats
  float* Dp = SS + NB1;                            // NB2 floats
  float* out = (float*)d_out;

  pd_stage1<<<NB1, 256, 0, stream>>>((const float4*)x, (float4*)P, SS);
  pd_stage2<<<NB2, 128, 0, stream>>>(P, Dp);
  pd_stage3<<<1, 256, 0, stream>>>(SS, Dp, out);
}